// DynamicUpsamplingFilter_15599321219373
// MI455X (gfx1250) — compile-verified
//
#include <hip/hip_runtime.h>
#include <hip/hip_bf16.h>
#include <stdint.h>
#include <stddef.h>

// Problem constants (n=1).
#define KH 5
#define KW 5
#define UPSQ 16
#define C 3
#define T 25
#define H 128
#define W 128
#define PW (W + KW - 1)        // 132 padded row width
#define PLANE (H * W)          // 16384
#define THW (T * H * W)        // 409600

typedef __attribute__((address_space(1))) int GASI;  // global int
typedef __attribute__((address_space(3))) int LASI;  // LDS int

__global__ __launch_bounds__(W) void
DynamicUpsamplingFilter_gfx1250_kernel(const float* __restrict__ x,
                                       const float* __restrict__ filters,
                                       float* __restrict__ out) {
    // Padded 5-row, 3-channel x tile for this (t,h): 3*5*132*4 = 7920 B LDS.
    __shared__ float sx[C][KH][PW];

    const int w = threadIdx.x;   // 0..127
    const int h = blockIdx.x;    // 0..127
    const int t = blockIdx.y;    // 0..24

    // ---- Stage x tile into LDS (async DMA path on gfx1250) ----------------
    const int E = C * KH * PW;   // 1980 elements
    for (int i = w; i < E; i += W) {
        const int c   = i / (KH * PW);
        const int rem = i % (KH * PW);
        const int dy  = rem / PW;
        const int col = rem % PW;
        const int r = h + dy - (KH / 2);
        const int g = col - (KW / 2);
        float* ldst = &sx[0][0][0] + i;
        if ((unsigned)r < (unsigned)H && (unsigned)g < (unsigned)W) {
            const float* gsrc = x + ((size_t)(c * T + t) * PLANE) + (size_t)r * W + g;
#if defined(__gfx1250__) && __has_builtin(__builtin_amdgcn_global_load_async_to_lds_b32)
            __builtin_amdgcn_global_load_async_to_lds_b32((GASI*)gsrc, (LASI*)ldst,
                                                          /*offset=*/0, /*cpol=*/0);
#else
            *ldst = *gsrc;
#endif
        } else {
            *ldst = 0.0f;   // zero padding (halo outside image)
        }
    }
#if defined(__gfx1250__) && __has_builtin(__builtin_amdgcn_s_wait_asynccnt)
    __builtin_amdgcn_s_wait_asynccnt(0);
#elif defined(__gfx1250__)
    asm volatile("s_wait_asynccnt 0" ::: "memory");
#endif
    __syncthreads();

    // ---- Per-pixel (3x25)@(25x16) matmul, filters streamed NT -------------
    const size_t fbase = (size_t)t * PLANE + (size_t)h * W + w;

    float acc[C][UPSQ];
#pragma unroll
    for (int c = 0; c < C; ++c)
#pragma unroll
        for (int u = 0; u < UPSQ; ++u) acc[c][u] = 0.0f;

#pragma unroll
    for (int a = 0; a < KH; ++a) {
#pragma unroll
        for (int b = 0; b < KW; ++b) {
            const int k = a * KW + b;
            // patch value x[h+a-2][w+b-2] == sx[c][a][w+b]  (zero padded)
            const float p0 = sx[0][a][w + b];
            const float p1 = sx[1][a][w + b];
            const float p2 = sx[2][a][w + b];
            const float* fp = filters + (size_t)k * UPSQ * THW + fbase;
#pragma unroll
            for (int u = 0; u < UPSQ; ++u) {
                // filters[(k*16+u)] plane, coalesced 512B/line across the block,
                // non-temporal: 655 MB streamed exactly once, keep it out of L2.
                const float f = __builtin_nontemporal_load(fp + (size_t)u * THW);
                acc[0][u] = fmaf(p0, f, acc[0][u]);
                acc[1][u] = fmaf(p1, f, acc[1][u]);
                acc[2][u] = fmaf(p2, f, acc[2][u]);
            }
        }
    }

    // ---- Write out (n, c*16+u, t, h, w), non-temporal ----------------------
#pragma unroll
    for (int c = 0; c < C; ++c) {
#pragma unroll
        for (int u = 0; u < UPSQ; ++u) {
            __builtin_nontemporal_store(acc[c][u],
                                        out + (size_t)(c * UPSQ + u) * THW + fbase);
        }
    }
}

extern "C" void kernel_launch(void* const* d_in, const int* in_sizes, int n_in,
                              void* d_out, int out_size, void* d_ws, size_t ws_size,
                              hipStream_t stream) {
    const float* x       = (const float*)d_in[0];  // (1,3,25,128,128)
    const float* filters = (const float*)d_in[1];  // (1,25,16,25,128,128)
    float* out           = (float*)d_out;          // (1,48,25,128,128)

    dim3 grid(H, T);   // one block per (h, t) row
    dim3 block(W);     // lane = w, 4 waves (wave32)
    DynamicUpsamplingFilter_gfx1250_kernel<<<grid, block, 0, stream>>>(x, filters, out);
}